// RiemannFormerAttention_44521630990753
// MI455X (gfx1250) — compile-verified
//
#include <hip/hip_runtime.h>
#include <hip/hip_bf16.h>

typedef __attribute__((ext_vector_type(2))) float v2f;
typedef __attribute__((ext_vector_type(8))) float v8f;

#define DIMX 512
#define NH 8
#define HD 64
#define BATCH 2
#define SEQ 2048
#define EXP_STEPS 20
#define LDP 68   // pitch: even (8B-aligned v2f frags), mult of 4 (16B async chunks),
                 // 68 mod 64 = 4 -> conflict-free fragment feeds

// ---------------- async global->LDS (ASYNCcnt) with safe fallback ----------
#if defined(__gfx1250__) && __has_builtin(__builtin_amdgcn_global_load_async_to_lds_b128)
#define HAVE_ASYNC 1
#else
#define HAVE_ASYNC 0
#endif

__device__ __forceinline__ void cp_async16(const float* g, float* l) {
#if HAVE_ASYNC
    typedef int v4i_ __attribute__((vector_size(16)));
    __builtin_amdgcn_global_load_async_to_lds_b128(
        (__attribute__((address_space(1))) v4i_*)(void*)g,
        (__attribute__((address_space(3))) v4i_*)l, 0, 0);
#else
    *(float4*)l = *(const float4*)g;
#endif
}

template <int N>
__device__ __forceinline__ void wait_async() {
#if HAVE_ASYNC
#if __has_builtin(__builtin_amdgcn_s_wait_asynccnt)
    __builtin_amdgcn_s_wait_asynccnt(N);
#else
    asm volatile("s_wait_asynccnt %0" ::"i"(N) : "memory");
#endif
#endif
}

// ---------------- WMMA f32 16x16x4 fragment helpers ------------------------
// A 16x4 (MxK): lanes 0-15 M=lane, lanes 16-31 M=lane-16; vgpr pair = K{+0,+1},
// K base offset 2 for upper half-wave.  Contiguous & 8B aligned -> ds_load_b64.
__device__ __forceinline__ v2f frag_a(const float* lds, int row0, int kk, int lane) {
    int m = row0 + (lane & 15);
    int k = kk + ((lane >> 4) << 1);
    return *(const v2f*)(lds + m * LDP + k);
}
// B 4x16 (KxN) stored "row-per-n": B[k][n] at lds[n*LDP + k]  -> contiguous v2f
__device__ __forceinline__ v2f frag_bT(const float* lds, int kk, int col0, int lane) {
    int n = col0 + (lane & 15);
    int k = kk + ((lane >> 4) << 1);
    return *(const v2f*)(lds + n * LDP + k);
}
// B 4x16 (KxN) stored row-major [k][n] (used where async DMA fixes the layout)
__device__ __forceinline__ v2f frag_b(const float* lds, int kk, int col0, int lane) {
    int n = col0 + (lane & 15);
    int k = kk + ((lane >> 4) << 1);
    v2f b;
    b.x = lds[k * LDP + n];
    b.y = lds[(k + 1) * LDP + n];
    return b;
}
__device__ __forceinline__ v8f wmma_f32(v2f a, v2f b, v8f c) {
    return __builtin_amdgcn_wmma_f32_16x16x4_f32(false, a, false, b, (short)0, c, false, false);
}
// C/D 16x16: vgpr v -> row = v + (lane>=16 ? 8 : 0), col = lane&15

// ============================================================================
// GEMM: out[n,m] = sum_k A[n,k]*W[m,k] + bias[m], async double-buffered panels.
// W panel stored untransposed (row-per-m == row-per-n of B) -> frag_bT works.
// ============================================================================
__global__ __launch_bounds__(128) void gemm_xWT(const float* __restrict__ A,
                                                const float* __restrict__ W,
                                                const float* __restrict__ bias,
                                                float* __restrict__ out,
                                                int out_bhld) {
    __shared__ float As[2][64 * LDP];
    __shared__ float Bs[2][64 * LDP];
    const int tid = threadIdx.x, lane = tid & 31, wave = tid >> 5;
    const int row0 = blockIdx.x * 64;
    const int col0 = blockIdx.y * 64;
    v8f acc[4] = {};

    auto issue = [&](int k0, int buf) {
        for (int i = tid; i < 1024; i += 128) {           // 64x64 panel = 1024 x 16B
            int r = i >> 4, c4 = (i & 15) << 2;
            cp_async16(&A[(size_t)(row0 + r) * DIMX + k0 + c4], &As[buf][r * LDP + c4]);
        }
        for (int i = tid; i < 1024; i += 128) {
            int m = i >> 4, c4 = (i & 15) << 2;
            cp_async16(&W[(size_t)(col0 + m) * DIMX + k0 + c4], &Bs[buf][m * LDP + c4]);
        }
    };

    issue(0, 0);
    for (int chunk = 0; chunk < 8; ++chunk) {
        const int buf = chunk & 1;
        if (chunk < 7) {
            issue((chunk + 1) * 64, buf ^ 1);   // prefetch next K-panel
            wait_async<16>();                   // retire the 16 older ops (this buf)
        } else {
            wait_async<0>();
        }
        __syncthreads();
        const int r0 = wave * 16;
        for (int ks = 0; ks < 16; ++ks) {
            v2f a = frag_a(As[buf], r0, ks * 4, lane);
            for (int ct = 0; ct < 4; ++ct) {
                v2f b = frag_bT(Bs[buf], ks * 4, ct * 16, lane);
                acc[ct] = wmma_f32(a, b, acc[ct]);
            }
        }
        __syncthreads();   // all waves done reading buf before it is re-targeted
    }
    for (int ct = 0; ct < 4; ++ct)
        for (int v = 0; v < 8; ++v) {
            int r = wave * 16 + v + ((lane >> 4) << 3);
            int n = row0 + r;
            int m = col0 + ct * 16 + (lane & 15);
            float val = acc[ct][v] + bias[m];
            if (out_bhld) {
                int b = n / SEQ, l = n % SEQ, h = m >> 6, dd = m & 63;
                out[(((size_t)b * NH + h) * SEQ + l) * HD + dd] = val;
            } else {
                out[(size_t)n * DIMX + m] = val;
            }
        }
}

// ============================================================================
// Per-(h,l): E = Taylor_20(exp(-p*X[h])) (WMMA chain, wave-private row strips,
// barrier-free inner loop), then Qr/Kr = e^{log_s/2} * E @ {Q,K}.
// M stored transposed (MsT[n][k]) so B-fragments are contiguous v2f loads.
// ============================================================================
__global__ __launch_bounds__(128) void riemann_transform(const int* __restrict__ positions,
                                                         const float* __restrict__ Xg,
                                                         const float* __restrict__ log_s,
                                                         const float* __restrict__ Q,
                                                         const float* __restrict__ K,
                                                         float* __restrict__ Qr,
                                                         float* __restrict__ Kr) {
    __shared__ float MsT[64 * LDP];  // MsT[n][k] = (-p*X)[k][n]
    __shared__ float Ts[64 * LDP];   // current Taylor term (row-major)
    __shared__ float Es[64 * LDP];   // accumulated exponential (row-major)
    const int tid = threadIdx.x, lane = tid & 31, wave = tid >> 5;
    const int l = blockIdx.x, h = blockIdx.y;
    const float p = (float)positions[l];
    const float* Xh = Xg + (size_t)h * HD * HD;
    for (int i = tid; i < 64 * 64; i += 128) {
        int k = i >> 6, n = i & 63;
        MsT[n * LDP + k] = -p * Xh[i];
        float id = (k == n) ? 1.0f : 0.0f;
        Ts[k * LDP + n] = id;
        Es[k * LDP + n] = id;
    }
    __syncthreads();
    const int r0 = wave * 16;
    for (int k = 1; k < EXP_STEPS; ++k) {
        const float inv_k = 1.0f / (float)k;
        v8f t[4] = {};
        for (int ks = 0; ks < 16; ++ks) {
            v2f a = frag_a(Ts, r0, ks * 4, lane);
            for (int ct = 0; ct < 4; ++ct) {
                v2f b = frag_bT(MsT, ks * 4, ct * 16, lane);
                t[ct] = wmma_f32(a, b, t[ct]);
            }
        }
        for (int ct = 0; ct < 4; ++ct)
            for (int v = 0; v < 8; ++v) {
                int r = r0 + v + ((lane >> 4) << 3);
                int c = ct * 16 + (lane & 15);
                float val = t[ct][v] * inv_k;
                Ts[r * LDP + c] = val;
                Es[r * LDP + c] += val;
            }
    }
    __syncthreads();
    const float s = __expf(0.5f * log_s[h]);   // Tinv = e^{log_s/2} * exp(-pX)
    {
        int b = tid >> 6, i = tid & 63;        // 128 threads == BATCH*HD
        size_t base = (((size_t)b * NH + h) * SEQ + l) * HD;
        float aq = 0.f, ak = 0.f;
        for (int j = 0; j < HD; ++j) {
            float e = Es[i * LDP + j];
            aq = fmaf(e, Q[base + j], aq);
            ak = fmaf(e, K[base + j], ak);
        }
        Qr[base + i] = s * aq;
        Kr[base + i] = s * ak;
    }
}

// ============================================================================
// Flash attention, async double-buffered K/V tiles, Gaussian log-space bias.
// K tile row-major == row-per-n of S's B operand -> contiguous frag_bT.
// ============================================================================
__global__ __launch_bounds__(128) void flash_attn(const float* __restrict__ Qr,
                                                  const float* __restrict__ Kr,
                                                  const float* __restrict__ V,
                                                  const int* __restrict__ positions,
                                                  const float* __restrict__ log_sigma,
                                                  float* __restrict__ Oout) {
    __shared__ float Qs[64 * LDP];
    __shared__ float Ps[64 * LDP];
    __shared__ float Ks[2][64 * LDP];
    __shared__ float Vs[2][64 * LDP];
    __shared__ float mrow[64], lrow[64], arow[64];
    const int tid = threadIdx.x, lane = tid & 31, wave = tid >> 5;
    const int qb = blockIdx.x, h = blockIdx.y, b = blockIdx.z;
    const size_t bh = ((size_t)b * NH + h) * SEQ;
    const float sigma = __expf(log_sigma[0]);
    const float inv2s2 = 1.0f / (2.0f * sigma * sigma);

    auto issueKV = [&](int jb, int buf) {
        for (int i = tid; i < 1024; i += 128) {
            int r = i >> 4, c4 = (i & 15) << 2;
            cp_async16(&Kr[(bh + jb * 64 + r) * HD + c4], &Ks[buf][r * LDP + c4]);
        }
        for (int i = tid; i < 1024; i += 128) {
            int r = i >> 4, c4 = (i & 15) << 2;
            cp_async16(&V[(bh + jb * 64 + r) * HD + c4], &Vs[buf][r * LDP + c4]);
        }
    };

    issueKV(0, 0);
    for (int i = tid; i < 1024; i += 128) {               // raw Q tile (scale folded later)
        int r = i >> 4, c4 = (i & 15) << 2;
        cp_async16(&Qr[(bh + qb * 64 + r) * HD + c4], &Qs[r * LDP + c4]);
    }
    if (tid < 64) { mrow[tid] = -3.0e38f; lrow[tid] = 0.f; }
    v8f o[4] = {};
    const int r0 = wave * 16;

    for (int jb = 0; jb < SEQ / 64; ++jb) {
        const int buf = jb & 1;
        wait_async<0>();       // only this iter's K/V (prefetched last iter) in flight
        __syncthreads();
        if (jb + 1 < SEQ / 64) issueKV(jb + 1, buf ^ 1);  // overlap with S/softmax/PV

        v8f st[4] = {};
        for (int ks = 0; ks < 16; ++ks) {                 // S = Q K^T
            v2f a = frag_a(Qs, r0, ks * 4, lane);
            for (int ct = 0; ct < 4; ++ct) {
                v2f bb = frag_bT(Ks[buf], ks * 4, ct * 16, lane);
                st[ct] = wmma_f32(a, bb, st[ct]);
            }
        }
        for (int ct = 0; ct < 4; ++ct)
            for (int v = 0; v < 8; ++v) {
                int r = r0 + v + ((lane >> 4) << 3);
                Ps[r * LDP + ct * 16 + (lane & 15)] = st[ct][v];
            }
        __syncthreads();
        if (tid < 64) {                                   // online softmax, 1 row/thread
            int r = tid;
            float qpos = (float)positions[qb * 64 + r];
            float mo = mrow[r], mx = mo;
            for (int c = 0; c < 64; ++c) {
                float diff = (float)positions[jb * 64 + c] - qpos;
                float bias = __logf(__expf(-diff * diff * inv2s2) + 1e-6f);
                float val = Ps[r * LDP + c] * 0.125f + bias;   // 1/sqrt(d) folded here
                Ps[r * LDP + c] = val;
                mx = fmaxf(mx, val);
            }
            float alpha = __expf(mo - mx);
            float sum = 0.f;
            for (int c = 0; c < 64; ++c) {
                float pe = __expf(Ps[r * LDP + c] - mx);
                Ps[r * LDP + c] = pe;
                sum += pe;
            }
            mrow[r] = mx; lrow[r] = lrow[r] * alpha + sum; arow[r] = alpha;
        }
        __syncthreads();
        {
            int rlo = r0 + ((lane >> 4) << 3);            // rescale O accumulator
            for (int ct = 0; ct < 4; ++ct)
                for (int v = 0; v < 8; ++v)
                    o[ct][v] *= arow[rlo + v];
        }
        for (int ks = 0; ks < 16; ++ks) {                 // O += P @ V (V row-major)
            v2f a = frag_a(Ps, r0, ks * 4, lane);
            for (int ct = 0; ct < 4; ++ct) {
                v2f bb = frag_b(Vs[buf], ks * 4, ct * 16, lane);
                o[ct] = wmma_f32(a, bb, o[ct]);
            }
        }
    }
    __syncthreads();
    int rlo = r0 + ((lane >> 4) << 3);
    for (int ct = 0; ct < 4; ++ct)
        for (int v = 0; v < 8; ++v) {
            int r = rlo + v;
            int c = ct * 16 + (lane & 15);
            int pos = qb * 64 + r;
            float val = o[ct][v] / lrow[r];
            Oout[(((size_t)b * SEQ + pos) * NH + h) * HD + c] = val;  // (B,L,D)
        }
}

// ============================================================================
extern "C" void kernel_launch(void* const* d_in, const int* in_sizes, int n_in,
                              void* d_out, int out_size, void* d_ws, size_t ws_size,
                              hipStream_t stream) {
    const float* x         = (const float*)d_in[0];
    const int*   positions = (const int*)d_in[1];
    const float* Wq = (const float*)d_in[2];
    const float* bq = (const float*)d_in[3];
    const float* Wk = (const float*)d_in[4];
    const float* bk = (const float*)d_in[5];
    const float* Wv = (const float*)d_in[6];
    const float* bv = (const float*)d_in[7];
    const float* Wo = (const float*)d_in[8];
    const float* bo = (const float*)d_in[9];
    const float* log_s     = (const float*)d_in[10];
    const float* X         = (const float*)d_in[11];
    const float* log_sigma = (const float*)d_in[12];
    float* out = (float*)d_out;

    const size_t per = (size_t)BATCH * NH * SEQ * HD;
    float* ws = (float*)d_ws;
    float* Q  = ws;
    float* K  = Q + per;
    float* Vt = K + per;
    float* Qr = Vt + per;
    float* Kr = Qr + per;
    float* AO = Kr + per;

    dim3 blk(128);
    dim3 gProj(BATCH * SEQ / 64, DIMX / 64);
    gemm_xWT<<<gProj, blk, 0, stream>>>(x, Wq, bq, Q, 1);
    gemm_xWT<<<gProj, blk, 0, stream>>>(x, Wk, bk, K, 1);
    gemm_xWT<<<gProj, blk, 0, stream>>>(x, Wv, bv, Vt, 1);
    riemann_transform<<<dim3(SEQ, NH), blk, 0, stream>>>(positions, X, log_s, Q, K, Qr, Kr);
    flash_attn<<<dim3(SEQ / 64, NH, BATCH), blk, 0, stream>>>(Qr, Kr, Vt, positions, log_sigma, AO);
    gemm_xWT<<<gProj, blk, 0, stream>>>(AO, Wo, bo, out, 0);
}